// ItemToItemScorer_57913339020024
// MI455X (gfx1250) — compile-verified
//
#include <hip/hip_runtime.h>

// ItemToItemScorer: per-edge scoring
//   out[e] = (dot(h[src[e]], h[dst[e]]) + bias[nid[src[e]]] + bias[nid[dst[e]]])
//            * (1/pop[src[e]]) * (1/pop[dst[e]])
//
// Strategy (MI455X / gfx1250, wave32):
//   One wave computes 16 edges. The 16 dot products are the DIAGONAL of
//   D = A x B where A = 16 src-rows (16xK) and B = 16 dst-rows as columns
//   (Kx16). We accumulate with V_WMMA_F32_16X16X4_F32 (full f32 precision,
//   K=4 per step, 32 steps for D_FEAT=128). FLOPs are free here (kernel is
//   an L2-resident gather problem: 51 MB table in 192 MB L2, 0.5 GB of
//   gathered reads), so the 16x "wasted" off-diagonal MACs cost nothing and
//   the WMMA unit replaces the whole cross-lane reduction tree.

typedef float v2f __attribute__((ext_vector_type(2)));
typedef float v8f __attribute__((ext_vector_type(8)));

#ifndef D_FEAT
#define D_FEAT 128
#endif

__global__ __launch_bounds__(256) void item2item_score_wmma(
    const float* __restrict__ h,     // [N_NODES, 128]
    const float* __restrict__ pop,   // [N_NODES, 1]
    const float* __restrict__ bias,  // [N_FULL]
    const int*   __restrict__ src,   // [E]
    const int*   __restrict__ dst,   // [E]
    const int*   __restrict__ nid,   // [N_NODES]
    float*       __restrict__ out,   // [E]
    int nEdges)
{
    const int lane = threadIdx.x & 31;
    const int wave = (int)((blockIdx.x * blockDim.x + threadIdx.x) >> 5);
    const int e0   = wave * 16;
    if (e0 >= nEdges) return;                 // wave-uniform

    const int m    = lane & 15;               // which of the 16 edges this lane helps with
    const int half = lane >> 4;               // 0: lanes 0-15, 1: lanes 16-31

    if (e0 + 16 <= nEdges) {                  // wave-uniform: full 16-edge tile
        const int e  = e0 + m;
        const int se = src[e];
        const int de = dst[e];
        const float* __restrict__ pA = h + (size_t)se * D_FEAT;   // A row (src)
        const float* __restrict__ pB = h + (size_t)de * D_FEAT;   // B col (dst)

        v8f c = {};
        #pragma unroll 4
        for (int k = 0; k < D_FEAT; k += 4) {
            // A 16x4 f32 layout: lanes 0-15 -> (K=k,k+1), lanes 16-31 -> (K=k+2,k+3)
            v2f a = *(const v2f*)(pA + k + 2 * half);
            // B 4x16 f32 layout: VGPR0 -> K row (k+half), VGPR1 -> K row (k+2+half), N = lane&15
            v2f b;
            b.x = pB[k + half];
            b.y = pB[k + 2 + half];
            c = __builtin_amdgcn_wmma_f32_16x16x4_f32(
                    /*neg_a=*/false, a, /*neg_b=*/false, b,
                    /*c_mod=*/(short)0, c, /*reuse_a=*/false, /*reuse_b=*/false);
        }

        // Extract diagonal D[M][M]:
        //   M in 0..7  -> vgpr M,   lane M
        //   M in 8..15 -> vgpr M-8, lane M+16
        float diag = 0.0f;
        #pragma unroll
        for (int i = 0; i < 8; ++i) {
            float vlo = __shfl(c[i], i);        // edge i
            float vhi = __shfl(c[i], i + 24);   // edge i+8 (vgpr i, lane i+24)
            diag = (lane == i)     ? vlo : diag;
            diag = (lane == i + 8) ? vhi : diag;
        }

        if (lane < 16) {                        // epilogue, divergence OK after WMMA
            const float bsum = bias[nid[se]] + bias[nid[de]];
            const float ipw  = (1.0f / pop[se]) * (1.0f / pop[de]);
            out[e] = (diag + bsum) * ipw;
        }
    } else {
        // Scalar tail (never taken for E % 16 == 0, kept for generality)
        const int e = e0 + lane;
        if (e < nEdges) {
            const int se = src[e], de = dst[e];
            const float* pA = h + (size_t)se * D_FEAT;
            const float* pB = h + (size_t)de * D_FEAT;
            float s = 0.0f;
            for (int k = 0; k < D_FEAT; ++k) s += pA[k] * pB[k];
            const float bsum = bias[nid[se]] + bias[nid[de]];
            const float ipw  = (1.0f / pop[se]) * (1.0f / pop[de]);
            out[e] = (s + bsum) * ipw;
        }
    }
}

extern "C" void kernel_launch(void* const* d_in, const int* in_sizes, int n_in,
                              void* d_out, int out_size, void* d_ws, size_t ws_size,
                              hipStream_t stream) {
    (void)n_in; (void)out_size; (void)d_ws; (void)ws_size;

    const float* h    = (const float*)d_in[0];   // [N_NODES*128]
    const float* pop  = (const float*)d_in[1];   // [N_NODES]
    const float* bias = (const float*)d_in[2];   // [N_FULL]
    const int*   src  = (const int*)d_in[3];     // [E]
    const int*   dst  = (const int*)d_in[4];     // [E]
    const int*   nid  = (const int*)d_in[5];     // [N_NODES]
    float*       out  = (float*)d_out;           // [E]

    const int nEdges  = in_sizes[3];

    const int waves   = (nEdges + 15) / 16;      // 16 edges per wave32
    const int threads = waves * 32;
    const int block   = 256;                     // 8 waves per block
    const int grid    = (threads + block - 1) / block;

    item2item_score_wmma<<<grid, block, 0, stream>>>(h, pop, bias, src, dst, nid,
                                                     out, nEdges);
}